// FCVI_Net_78864189489850
// MI455X (gfx1250) — compile-verified
//
#include <hip/hip_runtime.h>
#include <math.h>

// Problem constants (from reference): P=769 params, S*B = 16*2048 = 32768 columns.
#define PP   769
#define LDL  772          // padded L row stride (multiple of 4, zeros in pad)
#define NCOL 32768        // S*B
#define BMSK 2047         // B-1 for col -> batch index
#define KC   128          // K chunk staged in LDS
#define KST  132          // LDS row stride (132 % 64 == 4 -> conflict-free)

typedef __attribute__((ext_vector_type(2))) float v2f;
typedef __attribute__((ext_vector_type(8))) float v8f;

__device__ __forceinline__ v8f wmma_f32_16x16x4(v2f a, v2f b, v8f c) {
  // D = A(16x4 f32) * B(4x16 f32) + C(16x16 f32) — full fp32 matrix core op
  return __builtin_amdgcn_wmma_f32_16x16x4_f32(false, a, false, b, (short)0, c,
                                               false, false);
}

// ---------------------------------------------------------------------------
// Kernel 1: expand packed lower-triangular cov_vector into dense padded L.
// ---------------------------------------------------------------------------
__global__ __launch_bounds__(256) void build_L_kernel(const float* __restrict__ cov,
                                                      float* __restrict__ Lb) {
  int t = blockIdx.x * 256 + threadIdx.x;
  if (t >= PP * LDL) return;
  int i = t / LDL;
  int j = t - i * LDL;
  float v = 0.0f;
  if (j < PP && j <= i) {
    v = cov[i * (i + 1) / 2 + j];
    if (i == j) v = expf(v);
  }
  Lb[t] = v;
}

// ---------------------------------------------------------------------------
// Branch-free K phase: blocks TLO..5 are all active, emit (6-TLO) WMMAs/step.
// LDS block t lives at t*16*KST floats; lane reads row=sub, K-pair=2*half.
// ---------------------------------------------------------------------------
template <int TLO>
__device__ __forceinline__ void kphase(int kbeg, int kend, int kgbase,
                                       const float* __restrict__ zc,
                                       const float* __restrict__ sA,
                                       int half, int sub,
                                       v8f& c0, v8f& c1, v8f& c2,
                                       v8f& c3, v8f& c4, v8f& c5) {
  const int lbase = sub * KST + 2 * half;
  for (int kk = kbeg; kk < kend; kk += 4) {
    v2f bm;
    bm.x = zc[(size_t)(kgbase + kk + 2 * half) * NCOL];
    bm.y = zc[(size_t)(kgbase + kk + 2 * half + 1) * NCOL];
    const int ao = lbase + kk;
    if constexpr (TLO <= 0) { v2f a; a.x = sA[0 * 16 * KST + ao]; a.y = sA[0 * 16 * KST + ao + 1]; c0 = wmma_f32_16x16x4(a, bm, c0); }
    if constexpr (TLO <= 1) { v2f a; a.x = sA[1 * 16 * KST + ao]; a.y = sA[1 * 16 * KST + ao + 1]; c1 = wmma_f32_16x16x4(a, bm, c1); }
    if constexpr (TLO <= 2) { v2f a; a.x = sA[2 * 16 * KST + ao]; a.y = sA[2 * 16 * KST + ao + 1]; c2 = wmma_f32_16x16x4(a, bm, c2); }
    if constexpr (TLO <= 3) { v2f a; a.x = sA[3 * 16 * KST + ao]; a.y = sA[3 * 16 * KST + ao + 1]; c3 = wmma_f32_16x16x4(a, bm, c3); }
    if constexpr (TLO <= 4) { v2f a; a.x = sA[4 * 16 * KST + ao]; a.y = sA[4 * 16 * KST + ao + 1]; c4 = wmma_f32_16x16x4(a, bm, c4); }
    if constexpr (TLO <= 5) { v2f a; a.x = sA[5 * 16 * KST + ao]; a.y = sA[5 * 16 * KST + ao + 1]; c5 = wmma_f32_16x16x4(a, bm, c5); }
  }
}

// ---------------------------------------------------------------------------
// Kernel 2: fused  samples = mean + L @ Z  -> per-column 2-layer MLP fold.
// Each wave32 owns 16 columns; 8 waves/block -> 128 columns/block.
// Two m-blocks per pass (6 WMMA tiles share each B load; halves Z re-reads).
// Triangularity: K extents e0 / 256+e0 / 512+e0; phase sets are branch-free.
// ---------------------------------------------------------------------------
__global__ __launch_bounds__(256) void fcvi_main_kernel(
    const float* __restrict__ Lb, const float* __restrict__ z,
    const float* __restrict__ x, const float* __restrict__ mean,
    float* __restrict__ out) {
  __shared__ float sA[6 * 16 * KST];  // 50,688 bytes

  const int tid  = threadIdx.x;
  const int lane = tid & 31;
  const int wave = tid >> 5;
  const int sub  = lane & 15;  // output column in tile == A-matrix row in block
  const int half = lane >> 4;  // K-pair selector / C-row-half selector
  const int gcol = blockIdx.x * 128 + wave * 16 + sub;
  const float xv = x[gcol & BMSK];
  const float* __restrict__ zcol = z + gcol;

  float outpart = 0.0f;

  for (int mp = 0; mp < 8; ++mp) {
    const int r00 = 32 * mp, r01 = r00 + 16;       // W0 row pair
    const int rb0 = 256 + r00, rb1 = rb0 + 16;     // b0 row pair
    const int r10 = 512 + r00, r11 = r10 + 16;     // W1 row pair
    const int e0 = r00 + 32;                       // unified K extent (pair max)
    const int e1 = 512 + e0;
    const int rbase[6] = {r00, r01, rb0, rb1, r10, r11};
    const int bext[6]  = {e0, e0, 256 + e0, 256 + e0, e1, e1};

    v8f a0 = {0.f, 0.f, 0.f, 0.f, 0.f, 0.f, 0.f, 0.f};
    v8f a1 = a0, a2 = a0, a3 = a0, a4 = a0, a5 = a0;

    for (int kk0 = 0; kk0 < e1; kk0 += KC) {
      __syncthreads();  // protect previous chunk's readers
      // cooperative stage of L chunk into LDS (only still-active blocks)
#pragma unroll
      for (int t = 0; t < 6; ++t) {
        int e = bext[t] - kk0;
        if (e <= 0) continue;
        if (e > KC) e = KC;
#pragma unroll 4
        for (int r = 0; r < 16; ++r) {
          const float* src = Lb + (size_t)(rbase[t] + r) * LDL + kk0;
          float* dst = &sA[(t * 16 + r) * KST];
          for (int c = tid; c < e; c += 256) dst[c] = src[c];
        }
      }
      __syncthreads();

      int kend = e1 - kk0;
      if (kend > KC) kend = KC;
      // active-block set changes only at K = e0 and K = 256+e0 (>=256 apart),
      // so at most one transition per 128-wide chunk -> two branch-free phases
      const int tlo = (kk0 < e0) ? 0 : (kk0 < 256 + e0) ? 2 : 4;
      const int sbound = (tlo == 0) ? e0 : (tlo == 2) ? 256 + e0 : e1;
      int s = sbound - kk0;
      if (s > kend) s = kend;

      if (tlo == 0)      kphase<0>(0, s, kk0, zcol, sA, half, sub, a0, a1, a2, a3, a4, a5);
      else if (tlo == 2) kphase<2>(0, s, kk0, zcol, sA, half, sub, a0, a1, a2, a3, a4, a5);
      else               kphase<4>(0, s, kk0, zcol, sA, half, sub, a0, a1, a2, a3, a4, a5);
      if (s < kend) {
        if (tlo == 0) kphase<2>(s, kend, kk0, zcol, sA, half, sub, a0, a1, a2, a3, a4, a5);
        else          kphase<4>(s, kend, kk0, zcol, sA, half, sub, a0, a1, a2, a3, a4, a5);
      }
    }

    // Epilogue for these 32 hidden units: lane holds rows v+8*half, col=sub.
#pragma unroll
    for (int v = 0; v < 8; ++v) {
      const int rl = v + 8 * half;
      const float w0e = a0[v] + mean[r00 + rl];
      const float b0e = a2[v] + mean[rb0 + rl];
      const float w1e = a4[v] + mean[r10 + rl];
      outpart = fmaf(fmaxf(fmaf(xv, w0e, b0e), 0.0f), w1e, outpart);
      const float w0o = a1[v] + mean[r01 + rl];
      const float b0o = a3[v] + mean[rb1 + rl];
      const float w1o = a5[v] + mean[r11 + rl];
      outpart = fmaf(fmaxf(fmaf(xv, w0o, b0o), 0.0f), w1o, outpart);
    }
  }

  // b1 row (samples[768]): vector dot of L[768,:] with Z column, split by half.
  float bpart = (half == 0) ? mean[768] : 0.0f;
  const float* lrow = Lb + (size_t)768 * LDL;
  for (int j = half * 4; j < 768; j += 8) {
    const float4 lv = *(const float4*)(lrow + j);
    bpart = fmaf(lv.x, zcol[(size_t)(j + 0) * NCOL], bpart);
    bpart = fmaf(lv.y, zcol[(size_t)(j + 1) * NCOL], bpart);
    bpart = fmaf(lv.z, zcol[(size_t)(j + 2) * NCOL], bpart);
    bpart = fmaf(lv.w, zcol[(size_t)(j + 3) * NCOL], bpart);
  }
  if (half == 0) bpart = fmaf(lrow[768], zcol[(size_t)768 * NCOL], bpart);

  outpart += bpart;
  const float other = __shfl_xor(outpart, 16, 32);  // combine row-halves (wave32)
  if (half == 0) out[gcol] = outpart + other;
}

// ---------------------------------------------------------------------------
extern "C" void kernel_launch(void* const* d_in, const int* in_sizes, int n_in,
                              void* d_out, int out_size, void* d_ws, size_t ws_size,
                              hipStream_t stream) {
  (void)in_sizes; (void)n_in; (void)out_size; (void)ws_size;
  const float* x    = (const float*)d_in[0];  // (2048,)
  const float* mean = (const float*)d_in[1];  // (769,)
  const float* cov  = (const float*)d_in[2];  // (769*770/2,)
  const float* z    = (const float*)d_in[3];  // (769, 16, 2048) row-major
  float* outp = (float*)d_out;                // (16, 2048) flat
  float* Lbuf = (float*)d_ws;                 // needs 769*772*4 ≈ 2.27 MB

  build_L_kernel<<<(PP * LDL + 255) / 256, 256, 0, stream>>>(cov, Lbuf);
  fcvi_main_kernel<<<NCOL / 128, 256, 0, stream>>>(Lbuf, z, x, mean, outp);
}